// DftSpectrogram_19636590477809
// MI455X (gfx1250) — compile-verified
//
#include <hip/hip_runtime.h>
#include <math.h>

#define LEN     200
#define SHIFT_  150
#define FOUT    128          // only first NFFT/2 bins survive
#define BATCH   64
#define SLEN    480050       // 200 + 150*3199
#define TFRM    3200
#define TT      64           // frames per workgroup
#define NS      9650         // SHIFT_*(TT-1)+LEN samples staged in LDS
#define PITCH   129          // fft LDS pitch (avoid bank conflicts)
#define EPSF    1e-7f
#define INV_LN10 0.43429448190325176f

typedef __attribute__((ext_vector_type(2))) float v2f;
typedef __attribute__((ext_vector_type(8))) float v8f;

// ---------------- Pass 1: per-batch mean / 1/(std+eps) -------------------
__global__ __launch_bounds__(256) void dft_stats(const float* __restrict__ x,
                                                 float* __restrict__ ws) {
    __shared__ double sh[512];
    const int b   = blockIdx.x;
    const int tid = threadIdx.x;
    const float* xb = x + (size_t)b * SLEN;
    double s = 0.0, s2 = 0.0;
    for (int i = tid; i < SLEN; i += 256) {
        double v = (double)xb[i];
        s += v; s2 += v * v;
    }
    sh[tid] = s; sh[256 + tid] = s2;
    __syncthreads();
    for (int o = 128; o > 0; o >>= 1) {
        if (tid < o) { sh[tid] += sh[tid + o]; sh[256 + tid] += sh[256 + tid + o]; }
        __syncthreads();
    }
    if (tid == 0) {
        double m   = sh[0] / (double)SLEN;
        double var = sh[256] / (double)SLEN - m * m;
        if (var < 0.0) var = 0.0;
        ws[b]         = (float)m;
        ws[BATCH + b] = (float)(1.0 / (sqrt(var) + (double)EPSF));
    }
}

// ---------------- Pass 2: framed DFT (WMMA f32), log-mag, freq-normalize --
__global__ __launch_bounds__(256) void dft_spect(const float* __restrict__ x,
                                                 const float* __restrict__ kre,
                                                 const float* __restrict__ kim,
                                                 const float* __restrict__ ws,
                                                 float* __restrict__ out) {
    __shared__ float smem[NS];              // phase A: signal; phase B: fft[TT][PITCH]+mu+inv
    const int tid  = threadIdx.x;
    const int lane = tid & 31;
    const int wave = tid >> 5;              // 8 waves = 8 freq tiles of 16
    const int b    = blockIdx.y;
    const int t0   = blockIdx.x * TT;

    const float mu  = ws[b];
    const float inv = ws[BATCH + b];

    // ---- stage normalized signal window (covers 64 overlapping frames) ----
    const float* xb = x + (size_t)b * SLEN + (size_t)t0 * SHIFT_;
    for (int i = tid; i < NS; i += 256) smem[i] = (xb[i] - mu) * inv;
    __syncthreads();

    // ---- WMMA fragment geometry (ISA 7.12.2, 32-bit layouts) ----
    const int mrow = lane & 15;             // M (frame-in-tile) / N (freq-in-tile)
    const int hi   = lane >> 4;
    const int kb   = hi << 1;               // K sub-offset {0,2}
    const int f    = wave * 16 + mrow;      // freq bin 0..127

    const float* krp = kre + (size_t)f * LEN + kb;
    const float* kip = kim + (size_t)f * LEN + kb;

    const float* ap[4];
#pragma unroll
    for (int tt = 0; tt < 4; ++tt)
        ap[tt] = smem + SHIFT_ * (tt * 16 + mrow) + kb;

    v8f accR[4] = {}; v8f accI[4] = {};

#pragma unroll 2
    for (int k = 0; k < LEN; k += 4) {
        v2f br = *(const v2f*)(krp + k);    // B frag: L2-hot kernel matrix
        v2f bi = *(const v2f*)(kip + k);
#pragma unroll
        for (int tt = 0; tt < 4; ++tt) {
            v2f a = *(const v2f*)(ap[tt] + k);   // A frag from LDS
            accR[tt] = __builtin_amdgcn_wmma_f32_16x16x4_f32(
                false, a, false, br, (short)0, accR[tt], false, false);
            accI[tt] = __builtin_amdgcn_wmma_f32_16x16x4_f32(
                false, a, false, bi, (short)0, accI[tt], false, false);
        }
    }
    __syncthreads();                        // sig reads done; alias LDS for fft

    float* fftl = smem;                     // [TT][PITCH]
    float* mus  = smem + TT * PITCH;        // 8256
    float* invs = mus + TT;                 // 8320  (< NS, fits)

#pragma unroll
    for (int tt = 0; tt < 4; ++tt) {
#pragma unroll
        for (int r = 0; r < 8; ++r) {
            float re  = accR[tt][r];
            float im  = accI[tt][r];
            float mag = sqrtf(re * re + im * im);
            float v   = logf(fmaxf(mag, EPSF)) * INV_LN10;
            int   tl  = tt * 16 + r + 8 * hi;       // frame within chunk
            fftl[tl * PITCH + f] = v;
        }
    }
    __syncthreads();

    // ---- per-frame normalization over the 128 freq bins ----
    if (tid < TT) {
        float s = 0.f, s2 = 0.f;
        const float* row = fftl + tid * PITCH;
        for (int ff = 0; ff < FOUT; ++ff) { float v = row[ff]; s += v; s2 += v * v; }
        float m   = s * (1.0f / FOUT);
        float var = fmaxf(s2 * (1.0f / FOUT) - m * m, 0.0f);
        mus[tid]  = m;
        invs[tid] = 1.0f / (sqrtf(var) + EPSF);
    }
    __syncthreads();

    // ---- coalesced write: out[b][f][t], t fastest ----
    float* outb = out + (size_t)b * FOUT * TFRM;
    for (int i = tid; i < TT * FOUT; i += 256) {
        int ff = i >> 6;                    // /TT
        int tl = i & (TT - 1);
        float v = (fftl[tl * PITCH + ff] - mus[tl]) * invs[tl];
        outb[(size_t)ff * TFRM + (t0 + tl)] = v;
    }
}

extern "C" void kernel_launch(void* const* d_in, const int* in_sizes, int n_in,
                              void* d_out, int out_size, void* d_ws, size_t ws_size,
                              hipStream_t stream) {
    const float* x   = (const float*)d_in[0];
    const float* kre = (const float*)d_in[1];
    const float* kim = (const float*)d_in[2];
    float* out = (float*)d_out;
    float* ws  = (float*)d_ws;              // 128 floats: mu[64], inv[64]

    dft_stats<<<dim3(BATCH), dim3(256), 0, stream>>>(x, ws);
    dft_spect<<<dim3(TFRM / TT, BATCH), dim3(256), 0, stream>>>(x, kre, kim, ws, out);
}